// NavigationEnhacementLTC_53669911331212
// MI455X (gfx1250) — compile-verified
//
#include <hip/hip_runtime.h>
#include <hip/hip_bf16.h>

typedef float v8f __attribute__((ext_vector_type(8)));
typedef float v2f __attribute__((ext_vector_type(2)));

#define B_   128
#define T_   1024
#define D_   128
#define U_   128
#define R_   (B_ * T_)   // 131072 rows

__device__ __forceinline__ float softplusf(float x) {
    // log1p(exp(x)) with overflow guard (matches jax.nn.softplus numerically)
    return x > 20.0f ? x : log1pf(__expf(x));
}

__device__ __forceinline__ float fast_sigmoid(float z) {
    // sigmoid(z) = 1 / (1 + 2^(-z*log2(e)))  -> v_exp_f32 + v_rcp_f32
    float e = __builtin_amdgcn_exp2f(-1.44269504088896f * z);
    return __builtin_amdgcn_rcpf(1.0f + e);
}

// ---------------------------------------------------------------------------
// Kernel 1: h = LayerNorm(x @ pre_W^T + pre_b)   via V_WMMA_F32_16X16X4_F32
// One block = 16 rows x 128 cols. 8 waves, wave w computes N-tile [16w,16w+16).
// A (16x4 f32): lanes 0-15 hold M, vgpr0/1 = K0/K1; lanes 16-31 = K2/K3.
// B (4x16 f32): lanes hold N, mirrored K layout. B[k][n] = pre_W[n][k].
// ---------------------------------------------------------------------------
__global__ __launch_bounds__(256) void k_pre_gemm_ln(
    const float* __restrict__ x, const float* __restrict__ preW,
    const float* __restrict__ pre_b, const float* __restrict__ lng,
    const float* __restrict__ lnb, float* __restrict__ h)
{
    __shared__ float sX[16 * 128];
    __shared__ float sO[16 * 128];
    __shared__ float sMean[16];
    __shared__ float sRstd[16];

    const int tid  = threadIdx.x;
    const int wave = tid >> 5;
    const int lane = tid & 31;
    const int l15  = lane & 15;
    const int hi   = lane >> 4;
    const int row0 = blockIdx.x * 16;

    for (int i2 = tid; i2 < 16 * 128; i2 += 256)
        sX[i2] = x[row0 * 128 + i2];
    __syncthreads();

    const int col0 = wave * 16;
    v8f acc = {0.f, 0.f, 0.f, 0.f, 0.f, 0.f, 0.f, 0.f};
    const float* bp = preW + (col0 + l15) * 128 + 2 * hi;  // pre_W row = output col
    const float* ap = sX + l15 * 128 + 2 * hi;

    #pragma unroll 8
    for (int kk = 0; kk < 128; kk += 4) {
        v2f a; a.x = ap[kk]; a.y = ap[kk + 1];
        v2f b; b.x = bp[kk]; b.y = bp[kk + 1];
        acc = __builtin_amdgcn_wmma_f32_16x16x4_f32(
            false, a, false, b, (short)0, acc, false, false);
    }

    const float pb = pre_b[col0 + l15];
    #pragma unroll
    for (int r = 0; r < 8; ++r) {
        int m = r + 8 * hi;                 // C layout: vgpr r, lane<16 -> M=r, else M=r+8
        sO[m * 128 + col0 + l15] = acc[r] + pb;
    }
    __syncthreads();

    if (tid < 16) {
        float s = 0.f, s2 = 0.f;
        for (int c = 0; c < 128; ++c) { float v = sO[tid * 128 + c]; s += v; s2 += v * v; }
        float mean = s * (1.f / 128.f);
        float var  = s2 * (1.f / 128.f) - mean * mean;
        sMean[tid] = mean;
        sRstd[tid] = rsqrtf(var + 1e-5f);
    }
    __syncthreads();

    for (int i2 = tid; i2 < 16 * 128; i2 += 256) {
        int r = i2 >> 7, c = i2 & 127;
        h[row0 * 128 + i2] = (sO[i2] - sMean[r]) * sRstd[r] * lng[c] + lnb[c];
    }
}

// ---------------------------------------------------------------------------
// Kernel 2: pack params as float4 {sigma, mu, softplus(w), softplus(w)*erev}
// for both the sensory (DxU) and recurrent (UxU) synapse sets. Amortizes
// softplus once instead of per-timestep/per-block.
// ---------------------------------------------------------------------------
__global__ __launch_bounds__(256) void k_param_prep(
    const float* __restrict__ ss, const float* __restrict__ sm,
    const float* __restrict__ sw, const float* __restrict__ serev,
    const float* __restrict__ sg, const float* __restrict__ mu,
    const float* __restrict__ w,  const float* __restrict__ erev,
    float* __restrict__ pkS, float* __restrict__ pkR)
{
    int idx = blockIdx.x * 256 + threadIdx.x;
    if (idx >= D_ * U_) return;
    float swp = softplusf(sw[idx]);
    pkS[idx * 4 + 0] = ss[idx];
    pkS[idx * 4 + 1] = sm[idx];
    pkS[idx * 4 + 2] = swp;
    pkS[idx * 4 + 3] = swp * serev[idx];
    float wp = softplusf(w[idx]);
    pkR[idx * 4 + 0] = sg[idx];
    pkR[idx * 4 + 1] = mu[idx];
    pkR[idx * 4 + 2] = wp;
    pkR[idx * 4 + 3] = wp * erev[idx];
}

// ---------------------------------------------------------------------------
// Kernel 3: sensory sums, hoisted out of the scan (no v-dependence).
// One block = 8 (b,t) rows, 128 threads; thread i owns unit i, accumulates
// 8 rows in registers; param float4 loaded once per d and reused 8x.
// ---------------------------------------------------------------------------
__global__ __launch_bounds__(128) void k_sensory(
    const float* __restrict__ h, const float* __restrict__ pkS,
    const float* __restrict__ input_w, const float* __restrict__ input_b,
    float* __restrict__ wns, float* __restrict__ wds)
{
    __shared__ float sInp[8 * 128];
    const int tid  = threadIdx.x;
    const int row0 = blockIdx.x * 8;

    for (int k = tid; k < 8 * 128; k += 128) {
        int dcol = k & 127;
        sInp[k] = h[row0 * 128 + k] * input_w[dcol] + input_b[dcol];
    }
    __syncthreads();

    float num[8], den[8];
    #pragma unroll
    for (int r = 0; r < 8; ++r) { num[r] = 0.f; den[r] = 0.f; }

    const float4* pk = (const float4*)pkS;
    for (int d = 0; d < 128; ++d) {
        float4 p = pk[d * 128 + tid];
        #pragma unroll
        for (int r = 0; r < 8; ++r) {
            float z = p.x * (sInp[r * 128 + d] - p.y);
            float s = fast_sigmoid(z);
            num[r] = fmaf(p.w, s, num[r]);
            den[r] = fmaf(p.z, s, den[r]);
        }
    }
    #pragma unroll
    for (int r = 0; r < 8; ++r) {
        wns[(row0 + r) * 128 + tid] = num[r];
        wds[(row0 + r) * 128 + tid] = den[r];
    }
}

// ---------------------------------------------------------------------------
// Kernel 4: sequential LTC scan. One block per batch row; all recurrent
// params resident in 256KB of LDS (CDNA5 WGP LDS = 320KB). 512 threads:
// thread (q,i) reduces j in [32q, 32q+32) for unit i; LDS tree combine.
// ---------------------------------------------------------------------------
__global__ __launch_bounds__(512) void k_scan(
    const float* __restrict__ wns, const float* __restrict__ wds,
    const float* __restrict__ pkR,
    const float* __restrict__ gleak, const float* __restrict__ vleak,
    const float* __restrict__ cm, float* __restrict__ vlast, int T)
{
    extern __shared__ float lds[];
    float4* sP   = (float4*)lds;        // 16384 float4 = 256KB packed params
    float*  sV   = lds + 65536;         // 128
    float*  sNum = sV + 128;            // 512
    float*  sDen = sNum + 512;          // 512

    const int tid = threadIdx.x;
    const int i   = tid & 127;
    const int q   = tid >> 7;
    const int b   = blockIdx.x;

    const float4* gp = (const float4*)pkR;
    for (int k = tid; k < 16384; k += 512) sP[k] = gp[k];
    if (tid < 128) sV[tid] = 0.f;

    const float gl   = softplusf(gleak[i]);
    const float glvl = gl * vleak[i];
    const float ct   = softplusf(cm[i]) * 4.0f;   // cm / (1/ode_unfolds)
    const float cg   = ct + gl + 1e-8f;

    const float* pn = wns + (long)b * T * 128;
    const float* pd = wds + (long)b * T * 128;
    __syncthreads();

    const int j0 = q * 32;
    for (int t = 0; t < T; ++t) {
        float ns = 0.f, dsv = 0.f;
        if (q == 0) {
            ns  = pn[t * 128 + i];
            dsv = pd[t * 128 + i];
            if (t + 1 < T) {
                __builtin_prefetch(pn + (t + 1) * 128 + i, 0, 0);
                __builtin_prefetch(pd + (t + 1) * 128 + i, 0, 0);
            }
        }
        #pragma unroll 1
        for (int u = 0; u < 4; ++u) {
            float num = 0.f, den = 0.f;
            #pragma unroll 4
            for (int jj = 0; jj < 32; ++jj) {
                int j = j0 + jj;
                float  vj = sV[j];          // LDS broadcast
                float4 p  = sP[j * 128 + i];
                float z = p.x * (vj - p.y);
                float s = fast_sigmoid(z);
                num = fmaf(p.w, s, num);
                den = fmaf(p.z, s, den);
            }
            sNum[q * 128 + i] = num;
            sDen[q * 128 + i] = den;
            __syncthreads();
            if (q == 0) {
                float wn = sNum[i] + sNum[128 + i] + sNum[256 + i] + sNum[384 + i] + ns;
                float wd = sDen[i] + sDen[128 + i] + sDen[256 + i] + sDen[384 + i] + dsv;
                sV[i] = (ct * sV[i] + glvl + wn) * __builtin_amdgcn_rcpf(cg + wd);
            }
            __syncthreads();
        }
    }
    if (tid < 128) vlast[b * 128 + tid] = sV[tid];
}

// ---------------------------------------------------------------------------
// Kernel 5: output affine + LayerNorm + 15-way head projection. Tiny.
// ---------------------------------------------------------------------------
__global__ __launch_bounds__(128) void k_finalize(
    const float* __restrict__ vlast, const float* __restrict__ ow,
    const float* __restrict__ ob, const float* __restrict__ lng,
    const float* __restrict__ lnb, const float* __restrict__ hW,
    const float* __restrict__ hb, float* __restrict__ out)
{
    __shared__ float sO[128];
    __shared__ float stats[2];
    const int i = threadIdx.x;
    const int b = blockIdx.x;

    float v = vlast[b * 128 + i] * ow[i] + ob[i];
    sO[i] = v;
    __syncthreads();
    if (i == 0) {
        float s = 0.f, s2 = 0.f;
        for (int c = 0; c < 128; ++c) { float t = sO[c]; s += t; s2 += t * t; }
        float mean = s * (1.f / 128.f);
        float var  = s2 * (1.f / 128.f) - mean * mean;
        stats[0] = mean;
        stats[1] = rsqrtf(var + 1e-5f);
    }
    __syncthreads();
    float n = (v - stats[0]) * stats[1] * lng[i] + lnb[i];
    __syncthreads();
    sO[i] = n;
    __syncthreads();
    if (i < 15) {
        float acc = hb[i];
        for (int u = 0; u < 128; ++u) acc = fmaf(hW[i * 128 + u], sO[u], acc);
        out[b * 15 + i] = acc;
    }
}

// ---------------------------------------------------------------------------
extern "C" void kernel_launch(void* const* d_in, const int* in_sizes, int n_in,
                              void* d_out, int out_size, void* d_ws, size_t ws_size,
                              hipStream_t stream) {
    const float* x       = (const float*)d_in[0];
    const float* preW    = (const float*)d_in[1];
    const float* pre_b   = (const float*)d_in[2];
    const float* plg     = (const float*)d_in[3];
    const float* plb     = (const float*)d_in[4];
    const float* gleak   = (const float*)d_in[5];
    const float* vleak   = (const float*)d_in[6];
    const float* cm      = (const float*)d_in[7];
    const float* sigma   = (const float*)d_in[8];
    const float* mu      = (const float*)d_in[9];
    const float* w       = (const float*)d_in[10];
    const float* erev    = (const float*)d_in[11];
    const float* ss      = (const float*)d_in[12];
    const float* sm      = (const float*)d_in[13];
    const float* sw      = (const float*)d_in[14];
    const float* serev   = (const float*)d_in[15];
    const float* input_w = (const float*)d_in[16];
    const float* input_b = (const float*)d_in[17];
    const float* ow      = (const float*)d_in[18];
    const float* ob      = (const float*)d_in[19];
    const float* ltcg    = (const float*)d_in[20];
    const float* ltcb    = (const float*)d_in[21];
    const float* hW      = (const float*)d_in[22];
    const float* hb      = (const float*)d_in[23];

    float* ws    = (float*)d_ws;
    float* h     = ws;                    // R*D   = 16,777,216 f32
    float* wns   = h   + (size_t)R_ * D_; // R*U
    float* wds   = wns + (size_t)R_ * U_; // R*U
    float* pkS   = wds + (size_t)R_ * U_; // D*U*4 = 65536
    float* pkR   = pkS + (size_t)D_ * U_ * 4;
    float* vlast = pkR + (size_t)U_ * U_ * 4;

    k_pre_gemm_ln<<<R_ / 16, 256, 0, stream>>>(x, preW, pre_b, plg, plb, h);
    k_param_prep<<<(D_ * U_ + 255) / 256, 256, 0, stream>>>(
        ss, sm, sw, serev, sigma, mu, w, erev, pkS, pkR);
    k_sensory<<<R_ / 8, 128, 0, stream>>>(h, pkS, input_w, input_b, wns, wds);

    size_t smem = (size_t)(65536 + 128 + 512 + 512) * sizeof(float); // ~267KB < 320KB WGP LDS
    (void)hipFuncSetAttribute((const void*)k_scan,
                              hipFuncAttributeMaxDynamicSharedMemorySize, (int)smem);
    k_scan<<<B_, 512, smem, stream>>>(wns, wds, pkR, gleak, vleak, cm, vlast, T_);

    k_finalize<<<B_, 128, 0, stream>>>(vlast, ow, ob, ltcg, ltcb, hW, hb, (float*)d_out);
}